// SEFT_15985868276151
// MI455X (gfx1250) — compile-verified
//
#include <hip/hip_runtime.h>
#include <hip/hip_bf16.h>

// ---------------------------------------------------------------------------
// SEFT forward for MI455X (gfx1250, wave32).
//   Phase A: masked reduction over src [512, 4096, 72] -> out96 [4096, 96]
//   Phase B: small dense layers with V_WMMA_F32_16X16X4_F32 (f32-exact)
// ---------------------------------------------------------------------------

#define T_LEN   512
#define BATCH   4096
#define SRC_F   72
#define D_INP   36
#define D_PE    16

typedef float v2f __attribute__((ext_vector_type(2)));
typedef float v8f __attribute__((ext_vector_type(8)));

// ln(215)/7 : timescales[i] = 215^(i/7) = exp(i * LOG215_OVER7)
#define LOG215_OVER7 0.76723397f

// ---------------------------------------------------------------------------
// Kernel A: streaming reduction.
// Block = 256 threads, handles 16 consecutive batch rows across all T=512.
// thread -> (b_local = tid&15, t_lane = tid>>4); t strided by 16 (32 iters).
// Accumulators per thread: cnt, val, peT[16], fcnt[36]  (all f32, registers).
// LDS reduction over the 16 t-lanes, then cooperative finalize of out96.
// ---------------------------------------------------------------------------
__global__ __launch_bounds__(256) void seft_reduce_kernel(
    const float* __restrict__ src,     // [T, B, 72]
    const float* __restrict__ times,   // [T, B]
    const float* __restrict__ lv_w,    // [16]
    const float* __restrict__ lv_b,    // [16]
    float* __restrict__ out96)         // [B, 96]
{
  __shared__ float lds_acc[16][16][54];   // [b_local][t_lane][stat]
  __shared__ float lds_red[16][54];
  __shared__ float lds_pevar[D_INP * D_PE];

  const int tid = threadIdx.x;
  const int bl  = tid & 15;
  const int tl  = tid >> 4;
  const int b   = blockIdx.x * 16 + bl;

  float ts[8];
  #pragma unroll
  for (int i = 0; i < 8; ++i) ts[i] = __expf(LOG215_OVER7 * (float)i);

  // pe table for feature indices: lds_pevar[f*16 + d], d<8 -> sin, d>=8 -> cos
  for (int idx = tid; idx < D_INP * D_PE; idx += 256) {
    const int f = idx >> 4;
    const int d = idx & 15;
    const float s = (float)f / ts[d & 7];
    lds_pevar[idx] = (d < 8) ? __sinf(s) : __cosf(s);
  }

  float cnt = 0.0f, val = 0.0f;
  float peT[16];
  float fc[D_INP];
  #pragma unroll
  for (int i = 0; i < 16; ++i) peT[i] = 0.0f;
  #pragma unroll
  for (int f = 0; f < D_INP; ++f) fc[f] = 0.0f;

  for (int t = tl; t < T_LEN; t += 16) {
    const float* row = src + ((size_t)t * BATCH + b) * SRC_F;
    if (t + 16 < T_LEN) {
      // prefetch next iteration's row (gfx1250 global_prefetch path)
      __builtin_prefetch(row + (size_t)16 * BATCH * SRC_F, 0, 1);
    }
    float nz = 0.0f;
    #pragma unroll
    for (int q = 0; q < 9; ++q) {
      const float4 v = ((const float4*)row)[q];
      const float m0 = (v.x != 0.0f) ? 1.0f : 0.0f;
      const float m1 = (v.y != 0.0f) ? 1.0f : 0.0f;
      const float m2 = (v.z != 0.0f) ? 1.0f : 0.0f;
      const float m3 = (v.w != 0.0f) ? 1.0f : 0.0f;
      nz += m0 + m1 + m2 + m3;
      val += v.x + v.y + v.z + v.w;   // fea*mask == fea (zeros add zero)
      fc[q * 4 + 0] += m0;
      fc[q * 4 + 1] += m1;
      fc[q * 4 + 2] += m2;
      fc[q * 4 + 3] += m3;
    }
    cnt += nz;
    const float tv = times[(size_t)t * BATCH + b];
    #pragma unroll
    for (int i = 0; i < 8; ++i) {
      const float s = tv / ts[i];
      peT[i]     += __sinf(s) * nz;
      peT[i + 8] += __cosf(s) * nz;
    }
  }

  // stash accumulators
  {
    float* dst = &lds_acc[bl][tl][0];
    dst[0] = cnt;
    dst[1] = val;
    #pragma unroll
    for (int i = 0; i < 16; ++i) dst[2 + i] = peT[i];
    #pragma unroll
    for (int f = 0; f < D_INP; ++f) dst[18 + f] = fc[f];
  }
  __syncthreads();

  // reduce over the 16 t-lanes (fixed order -> deterministic)
  for (int idx = tid; idx < 16 * 54; idx += 256) {
    const int bb = idx / 54;
    const int c  = idx % 54;
    float s = 0.0f;
    #pragma unroll
    for (int j = 0; j < 16; ++j) s += lds_acc[bb][j][c];
    lds_red[bb][c] = s;
  }
  __syncthreads();

  // finalize: f_prime[48] per row, duplicated into out96 (zero if cnt==0)
  for (int idx = tid; idx < 16 * 48; idx += 256) {
    const int bb = idx / 48;
    const int c  = idx % 48;
    const float cntb = lds_red[bb][0];
    const float safe = fmaxf(cntb, 1.0f);
    float v;
    if (c < 16) {                       // pe_mean
      v = lds_red[bb][2 + c] / safe;
    } else if (c < 32) {                // values_mean = val_mean*lv_w + lv_b
      const int j = c - 16;
      v = (lds_red[bb][1] / safe) * lv_w[j] + lv_b[j];
    } else {                            // var_mean: pe_var^T @ f_cnt / safe
      const int d = c - 32;
      float s = 0.0f;
      #pragma unroll
      for (int f = 0; f < D_INP; ++f)
        s += lds_pevar[f * D_PE + d] * lds_red[bb][18 + f];
      v = s / safe;
    }
    if (!(cntb > 0.0f)) v = 0.0f;
    float* o = out96 + (size_t)(blockIdx.x * 16 + bb) * 96;
    o[c]      = v;
    o[c + 48] = v;
  }
}

// ---------------------------------------------------------------------------
// WMMA helpers: one wave (32 lanes) computes a 16x16 f32 tile via
// V_WMMA_F32_16X16X4_F32, K stepped by 4.
// A (16x4) layout: lanes 0-15 -> M=lane, K={k0,k0+1}; lanes 16-31 -> K={k0+2,k0+3}
// B (4x16) layout: lanes 0-15 -> N=lane, K={k0,k0+1}; lanes 16-31 -> K={k0+2,k0+3}
// C/D (16x16): VGPR r -> M = r + (lane>=16 ? 8 : 0), N = lane&15
// ---------------------------------------------------------------------------
__device__ __forceinline__ v8f wmma_tile_f32(
    const float* __restrict__ A, int lda, int row0,   // A row-major [M, lda]
    const float* __restrict__ W, int ldw, int n0,     // W row-major [N, ldw]; B[k][n]=W[n0+n][k]
    int K, v8f acc, int lane)
{
  const int half = lane >> 4;       // 0 or 1
  const int lid  = lane & 15;
  const int ks   = half * 2;
  for (int k0 = 0; k0 < K; k0 += 4) {
    v2f a, bf;
    const float* ap = A + (size_t)(row0 + lid) * lda + k0 + ks;
    a.x  = ap[0];
    a.y  = ap[1];
    const float* bp = W + (size_t)(n0 + lid) * ldw + k0 + ks;
    bf.x = bp[0];
    bf.y = bp[1];
    acc = __builtin_amdgcn_wmma_f32_16x16x4_f32(false, a, false, bf,
                                                (short)0, acc, false, false);
  }
  return acc;
}

// ---------------------------------------------------------------------------
// Kernel B1: x[:, :128] = out96 @ lm_w^T + lm_b  (WMMA, M=4096 N=128 K=96)
//            x[:, 128:144] = static @ emb_w^T + emb_b  (VALU, K=9)
// Grid: (256, 9), 32 threads (1 wave) per block.
// ---------------------------------------------------------------------------
__global__ __launch_bounds__(32) void seft_stage1_kernel(
    const float* __restrict__ out96,    // [B, 96]
    const float* __restrict__ lm_w,     // [128, 96]
    const float* __restrict__ lm_b,     // [128]
    const float* __restrict__ stat,     // [B, 9]
    const float* __restrict__ emb_w,    // [16, 9]
    const float* __restrict__ emb_b,    // [16]
    float* __restrict__ x)              // [B, 144]
{
  const int lane  = threadIdx.x;
  const int row0  = blockIdx.x * 16;
  const int ntile = blockIdx.y;

  if (ntile < 8) {
    const int n0  = ntile * 16;
    const int col = n0 + (lane & 15);
    v8f acc;
    const float bias = lm_b[col];
    #pragma unroll
    for (int r = 0; r < 8; ++r) acc[r] = bias;
    acc = wmma_tile_f32(out96, 96, row0, lm_w, 96, n0, 96, acc, lane);
    const int mbase = row0 + (lane >> 4) * 8;
    #pragma unroll
    for (int r = 0; r < 8; ++r)
      x[(size_t)(mbase + r) * 144 + col] = acc[r];
  } else {
    // static embedding columns 128..143
    #pragma unroll
    for (int i = 0; i < 8; ++i) {
      const int idx = lane * 8 + i;          // 0..255
      const int bb  = idx >> 4;
      const int c   = idx & 15;
      float v = emb_b[c];
      #pragma unroll
      for (int k = 0; k < 9; ++k)
        v += stat[(size_t)(row0 + bb) * 9 + k] * emb_w[c * 9 + k];
      x[(size_t)(row0 + bb) * 144 + 128 + c] = v;
    }
  }
}

// ---------------------------------------------------------------------------
// Kernel B2: h = relu(x @ mlp1_w^T + mlp1_b)  (WMMA, M=4096 N=144 K=144)
// Grid: (256, 9), 32 threads per block.
// ---------------------------------------------------------------------------
__global__ __launch_bounds__(32) void seft_mlp1_kernel(
    const float* __restrict__ x,        // [B, 144]
    const float* __restrict__ mlp1_w,   // [144, 144]
    const float* __restrict__ mlp1_b,   // [144]
    float* __restrict__ h)              // [B, 144]
{
  const int lane = threadIdx.x;
  const int row0 = blockIdx.x * 16;
  const int n0   = blockIdx.y * 16;
  const int col  = n0 + (lane & 15);

  v8f acc;
  const float bias = mlp1_b[col];
  #pragma unroll
  for (int r = 0; r < 8; ++r) acc[r] = bias;
  acc = wmma_tile_f32(x, 144, row0, mlp1_w, 144, n0, 144, acc, lane);

  const int mbase = row0 + (lane >> 4) * 8;
  #pragma unroll
  for (int r = 0; r < 8; ++r)
    h[(size_t)(mbase + r) * 144 + col] = fmaxf(acc[r], 0.0f);
}

// ---------------------------------------------------------------------------
// Kernel B3: y = h @ mlp2_w^T + mlp2_b  ([B,144] x [2,144] -> [B,2], VALU)
// ---------------------------------------------------------------------------
__global__ __launch_bounds__(256) void seft_mlp2_kernel(
    const float* __restrict__ h,        // [B, 144]
    const float* __restrict__ mlp2_w,   // [2, 144]
    const float* __restrict__ mlp2_b,   // [2]
    float* __restrict__ out)            // [B, 2]
{
  const int b = blockIdx.x * blockDim.x + threadIdx.x;
  if (b >= BATCH) return;
  float a0 = mlp2_b[0];
  float a1 = mlp2_b[1];
  const float* hp = h + (size_t)b * 144;
  #pragma unroll 4
  for (int k = 0; k < 144; ++k) {
    const float hv = hp[k];
    a0 += hv * mlp2_w[k];
    a1 += hv * mlp2_w[144 + k];
  }
  out[(size_t)b * 2 + 0] = a0;
  out[(size_t)b * 2 + 1] = a1;
}

// ---------------------------------------------------------------------------
extern "C" void kernel_launch(void* const* d_in, const int* in_sizes, int n_in,
                              void* d_out, int out_size, void* d_ws, size_t ws_size,
                              hipStream_t stream) {
  const float* src    = (const float*)d_in[0];   // [512, 4096, 72]
  const float* stat   = (const float*)d_in[1];   // [4096, 9]
  const float* times  = (const float*)d_in[2];   // [512, 4096]
  // d_in[3] = lengths (unused by forward)
  const float* lv_w   = (const float*)d_in[4];   // [16,1]
  const float* lv_b   = (const float*)d_in[5];   // [16]
  const float* lm_w   = (const float*)d_in[6];   // [128,96]
  const float* lm_b   = (const float*)d_in[7];   // [128]
  const float* emb_w  = (const float*)d_in[8];   // [16,9]
  const float* emb_b  = (const float*)d_in[9];   // [16]
  const float* mlp1_w = (const float*)d_in[10];  // [144,144]
  const float* mlp1_b = (const float*)d_in[11];  // [144]
  const float* mlp2_w = (const float*)d_in[12];  // [2,144]
  const float* mlp2_b = (const float*)d_in[13];  // [2]

  float* ws    = (float*)d_ws;
  float* out96 = ws;                                // [4096,  96]
  float* x144  = out96 + (size_t)BATCH * 96;        // [4096, 144]
  float* h144  = x144  + (size_t)BATCH * 144;       // [4096, 144]
  float* out   = (float*)d_out;                     // [4096,   2]

  seft_reduce_kernel<<<BATCH / 16, 256, 0, stream>>>(src, times, lv_w, lv_b, out96);
  seft_stage1_kernel<<<dim3(BATCH / 16, 9), 32, 0, stream>>>(out96, lm_w, lm_b,
                                                             stat, emb_w, emb_b, x144);
  seft_mlp1_kernel<<<dim3(BATCH / 16, 9), 32, 0, stream>>>(x144, mlp1_w, mlp1_b, h144);
  seft_mlp2_kernel<<<(BATCH + 255) / 256, 256, 0, stream>>>(h144, mlp2_w, mlp2_b, out);
}